// MultiStep_57526791962975
// MI455X (gfx1250) — compile-verified
//
#include <hip/hip_runtime.h>
#include <stdint.h>

#define NSTEP   8
#define T_LEN   8192
#define TILE    4096
#define BLOCK   256
#define EPT     8                 // contiguous elements per thread per group
#define GROUPS  (TILE / (BLOCK * EPT))   // = 2

typedef unsigned int u32x4 __attribute__((ext_vector_type(4)));
typedef int          i32x8 __attribute__((ext_vector_type(8)));
typedef int          i32x4 __attribute__((ext_vector_type(4)));

// Low 32 bits of a flat pointer into LDS == LDS byte offset (CDNA5 aperture rule).
__device__ __forceinline__ unsigned lds_off(const void* p) {
    return (unsigned)(unsigned long long)(uintptr_t)p;
}

// 1D TDM tile load via Tensor Data Mover. Elements past valid_elems are OOB in
// tensor_dim0 and read as ZERO -> free halo zero-fill at row end.
__device__ __forceinline__ void tdm_load_1d(unsigned lds_addr, unsigned long long ga,
                                            unsigned valid_elems, unsigned tile_elems,
                                            unsigned data_size_code) {
    u32x4 g0 = { 1u,                                   // count=1
                 lds_addr,
                 (unsigned)(ga & 0xFFFFFFFFu),
                 (unsigned)((ga >> 32) & 0x01FFFFFFu) | (2u << 30) };  // type=2
    i32x8 g1 = { (int)((data_size_code & 3u) << 16),   // mask=0, data_size
                 (int)((valid_elems & 0xFFFFu) << 16), // tensor_dim0 lo
                 (int)(((valid_elems >> 16) & 0xFFFFu) | (1u << 16)),  // dim0 hi | tensor_dim1=1
                 (int)((tile_elems & 0xFFFFu) << 16),  // tile_dim0
                 0,                                    // tile_dim1 = tile_dim2 = 0 (1D tile)
                 (int)valid_elems,                     // tensor_dim0_stride
                 0, 0 };
    i32x4 g2 = {0, 0, 0, 0};
    i32x4 g3 = {0, 0, 0, 0};
#if defined(__clang_major__) && (__clang_major__ >= 23)
    i32x8 g4 = {0, 0, 0, 0, 0, 0, 0, 0};
    __builtin_amdgcn_tensor_load_to_lds(g0, g1, g2, g3, g4, 0);
#else
    __builtin_amdgcn_tensor_load_to_lds(g0, g1, g2, g3, 0);
#endif
}

// Collapse each byte of x to "any bit set", pack the four byte-flags into bits [3:0].
__device__ __forceinline__ unsigned pack_done4(unsigned x) {
    x |= x >> 4; x |= x >> 2; x |= x >> 1;
    x &= 0x01010101u;
    return (x & 1u) | ((x >> 7) & 2u) | ((x >> 14) & 4u) | ((x >> 21) & 8u);
}

__global__ __launch_bounds__(BLOCK)
void nstep_return_kernel(const float* __restrict__ reward,
                         const unsigned char* __restrict__ done,
                         float* __restrict__ out, int B) {
    __shared__ __align__(16) float         s_r[TILE + NSTEP + 8];  // pad for b128 tail reads
    __shared__ __align__(16) unsigned char s_d[TILE + NSTEP + 8];

    const int tilesPerRow = T_LEN / TILE;
    const int b      = (int)blockIdx.x / tilesPerRow;
    const int tStart = ((int)blockIdx.x % tilesPerRow) * TILE;
    const size_t rowBase = (size_t)b * (size_t)T_LEN;
    const unsigned validElems = (unsigned)(T_LEN - tStart);

    // Wave 0 drives the TDM for both tiles, then waits TENSORcnt.
    if (threadIdx.x < 32u) {
        tdm_load_1d(lds_off(&s_r[0]),
                    (unsigned long long)(uintptr_t)(reward + rowBase + (size_t)tStart),
                    validElems, TILE + NSTEP + 8, 2u);
        tdm_load_1d(lds_off(&s_d[0]),
                    (unsigned long long)(uintptr_t)(done + rowBase + (size_t)tStart),
                    validElems, TILE + NSTEP + 8, 0u);
        __builtin_amdgcn_s_wait_tensorcnt(0);
    }
    __syncthreads();

    constexpr float GP[NSTEP + 1] = {
        1.0f, 0.99f, 0.9801f, 0.970299f, 0.96059601f,
        0.9509900499f, 0.941480149401f, 0.93206534790699f, 0.9227446944279201f };
    constexpr float G9 = 0.9135172474836409f;          // gamma^9

    float* __restrict__ out_sum = out;
    float* __restrict__ out_tto = out + (size_t)B * (size_t)T_LEN;

#pragma unroll
    for (int k = 0; k < GROUPS; ++k) {
        const int loc = k * (BLOCK * EPT) + (int)threadIdx.x * EPT;  // multiple of 8
        const int gt0 = tStart + loc;

        // 16 rewards cover all eight overlapping 9-wide windows (max index 7+8=15).
        const float4 r0 = *(const float4*)&s_r[loc];
        const float4 r1 = *(const float4*)&s_r[loc + 4];
        const float4 r2 = *(const float4*)&s_r[loc + 8];
        const float4 r3 = *(const float4*)&s_r[loc + 12];
        const float r[16] = { r0.x, r0.y, r0.z, r0.w,  r1.x, r1.y, r1.z, r1.w,
                              r2.x, r2.y, r2.z, r2.w,  r3.x, r3.y, r3.z, r3.w };

        // 16 done flags -> 16-bit stop mask (8B-aligned b64 loads).
        const uint2 da = *(const uint2*)&s_d[loc];
        const uint2 db = *(const uint2*)&s_d[loc + 8];
        const unsigned m = pack_done4(da.x)
                         | (pack_done4(da.y) << 4)
                         | (pack_done4(db.x) << 8)
                         | (pack_done4(db.y) << 12);

        // Per-element 9-bit stop window (incl. implicit done at T-1) and cutoff L.
        const int e0 = (T_LEN - 1) - gt0;
        unsigned wm[EPT];
        int L[EPT];
#pragma unroll
        for (int j = 0; j < EPT; ++j) {
            unsigned w = (m >> j) & 0x1FFu;
            if ((unsigned)(e0 - j) < 9u) w |= 1u << (e0 - j);
            wm[j] = w;
            L[j] = __builtin_ctz(w | 0x100u);           // min(first stop offset, 8)
        }

        // Rightmost element: direct masked 9-term dot.
        float sLast = 0.0f;
#pragma unroll
        for (int i = 0; i <= NSTEP; ++i)
            sLast += (i <= L[EPT - 1]) ? GP[i] * r[(EPT - 1) + i] : 0.0f;

        // Elements EPT-2..0 via exact recurrence:
        //   stop at 0   -> summed = r[j]
        //   L[j+1] <  8 -> summed = r[j] + g*summed[j+1]
        //   L[j+1] == 8 -> summed = r[j] + g*summed[j+1] - g^9*r[j+9]
        float sums[EPT];
        sums[EPT - 1] = sLast;
        float prev = sLast;
#pragma unroll
        for (int j = EPT - 2; j >= 0; --j) {
            float c = fmaf(0.99f, prev, r[j]);
            const float t9 = (L[j + 1] == NSTEP) ? r[j + 9] : 0.0f;
            c = fmaf(-G9, t9, c);
            c = (wm[j] & 1u) ? r[j] : c;
            sums[j] = c;
            prev = c;
        }

        const float4 sv0 = { sums[0], sums[1], sums[2], sums[3] };
        const float4 sv1 = { sums[4], sums[5], sums[6], sums[7] };
        const float4 tv0 = { (float)L[0], (float)L[1], (float)L[2], (float)L[3] };
        const float4 tv1 = { (float)L[4], (float)L[5], (float)L[6], (float)L[7] };

        const size_t idx = rowBase + (size_t)gt0;       // 32B-aligned
        *(float4*)&out_sum[idx]     = sv0;
        *(float4*)&out_sum[idx + 4] = sv1;
        *(float4*)&out_tto[idx]     = tv0;
        *(float4*)&out_tto[idx + 4] = tv1;
    }
}

extern "C" void kernel_launch(void* const* d_in, const int* in_sizes, int n_in,
                              void* d_out, int out_size, void* d_ws, size_t ws_size,
                              hipStream_t stream) {
    (void)n_in; (void)out_size; (void)d_ws; (void)ws_size;
    const float*         reward = (const float*)d_in[0];
    const unsigned char* done   = (const unsigned char*)d_in[1];  // JAX bool = 1 byte
    float* out = (float*)d_out;

    const int B = in_sizes[0] / T_LEN;
    const int blocks = B * (T_LEN / TILE);
    nstep_return_kernel<<<dim3(blocks), dim3(BLOCK), 0, stream>>>(reward, done, out, B);
}